// DownBlock_66073776881956
// MI455X (gfx1250) — compile-verified
//
#include <hip/hip_runtime.h>

typedef float v2f __attribute__((ext_vector_type(2)));
typedef float v4f __attribute__((ext_vector_type(4)));
typedef float v8f __attribute__((ext_vector_type(8)));

#define COUT 64
#define BN_EPS 1e-5f

// ---------------------------------------------------------------- utilities
__global__ void zero_f32v4(v4f* __restrict__ p, int n4) {
  int i = blockIdx.x * blockDim.x + threadIdx.x;
  int s = gridDim.x * blockDim.x;
  v4f z = {};
  for (; i < n4; i += s) p[i] = z;
}

// ------------------------------------------------- gather-GEMM-scatter conv
// grid = (ceil(mtiles/8), K, 4); block = 256 (8 waves).
// One wave = one 16(pairs) x 16(Cout) output tile for one kernel offset k.
// fp32 WMMA: V_WMMA_F32_16X16X4_F32, fully unrolled over Cin in steps of 4.
template <int CIN>
__global__ void sparse_conv_wmma(const float* __restrict__ feats,
                                 const float* __restrict__ W,   // [K][CIN][64]
                                 const int*   __restrict__ pin, // [K][P]
                                 const int*   __restrict__ pout,// [K][P]
                                 float*       __restrict__ out, // [n_out][64]
                                 int P, int mtiles) {
  const int wave  = threadIdx.x >> 5;
  const int mtile = blockIdx.x * 8 + wave;
  if (mtile >= mtiles) return;               // wave-uniform: EXEC stays all-1

  const int k     = blockIdx.y;              // block-uniform (SGPR)
  const int ntile = blockIdx.z;              // block-uniform (SGPR)
  const int lane  = threadIdx.x & 31;
  const int m0    = mtile * 16;
  const int n0    = ntile * 16;
  const int half  = lane >> 4;               // 0: lanes 0-15, 1: lanes 16-31
  const int nl    = lane & 15;

  // A gather row for this lane (rows of the 16x4 A tile live in lanes 0..15/16..31)
  const int pidx = m0 + nl;
  const int arow = (pidx < P) ? pin[k * P + pidx] : 0;
  const float* aptr = feats + (long)arow * CIN + 2 * half;          // cols c0+2h, c0+2h+1
  const float* bptr = W + (long)k * CIN * COUT + 2 * half * COUT + n0 + nl;

  v8f c = {};
#pragma unroll
  for (int c0 = 0; c0 < CIN; c0 += 4) {
    v2f a = *reinterpret_cast<const v2f*>(aptr + c0);   // 8B-aligned b64 gather
    v2f b;
    b.x = bptr[c0 * COUT];                              // K-row c0+2h
    b.y = bptr[c0 * COUT + COUT];                       // K-row c0+2h+1
    c = __builtin_amdgcn_wmma_f32_16x16x4_f32(false, a, false, b,
                                              (short)0, c, false, false);
  }

  // Scatter: D VGPR r holds output row (r + 8*half), column n0+nl.
#pragma unroll
  for (int r = 0; r < 8; ++r) {
    const int pi = m0 + r + 8 * half;
    if (pi < P) {
      const int orow = pout[k * P + pi];
      atomicAdd(&out[(long)orow * COUT + n0 + nl], c[r]);
    }
  }
}

// ------------------------------------------------------------ BN statistics
__global__ void col_stats(const float* __restrict__ x, int n_elems,
                          float* __restrict__ stats) {  // stats[0:64]=sum, [64:128]=sumsq
  __shared__ float ssum[COUT];
  __shared__ float ssq[COUT];
  if (threadIdx.x < COUT) { ssum[threadIdx.x] = 0.f; ssq[threadIdx.x] = 0.f; }
  __syncthreads();
  int idx    = blockIdx.x * blockDim.x + threadIdx.x;
  int stride = gridDim.x * blockDim.x;       // multiple of 64 -> channel invariant
  const int ch = idx & (COUT - 1);
  float lsum = 0.f, lsq = 0.f;
  for (int i = idx; i < n_elems; i += stride) {
    float v = x[i];
    lsum += v;
    lsq  += v * v;
  }
  atomicAdd(&ssum[ch], lsum);
  atomicAdd(&ssq[ch], lsq);
  __syncthreads();
  if (threadIdx.x < COUT) {
    atomicAdd(&stats[threadIdx.x], ssum[threadIdx.x]);
    atomicAdd(&stats[COUT + threadIdx.x], ssq[threadIdx.x]);
  }
}

__global__ void bn_finalize(const float* __restrict__ stats,
                            const float* __restrict__ g,
                            const float* __restrict__ b,
                            float inv_n, float* __restrict__ ab) {
  int c = threadIdx.x;
  if (c < COUT) {
    float mean = stats[c] * inv_n;
    float var  = stats[COUT + c] * inv_n - mean * mean;   // biased variance
    float s    = g[c] * rsqrtf(var + BN_EPS);
    ab[c]        = s;
    ab[COUT + c] = b[c] - mean * s;
  }
}

__global__ void bn_apply(const float* __restrict__ y,
                         const float* __restrict__ ab,
                         float* __restrict__ x, int n_elems) {
  int i = blockIdx.x * blockDim.x + threadIdx.x;
  int s = gridDim.x * blockDim.x;
  for (; i < n_elems; i += s) {
    int ch = i & (COUT - 1);
    float v = y[i] * ab[ch] + ab[COUT + ch];
    x[i] = v > 0.f ? v : 0.f;
  }
}

// Final: out = x_net + relu(bn(y2))
__global__ void bn_apply_add(const float* __restrict__ y,
                             const float* __restrict__ ab,
                             const float* __restrict__ xnet,
                             float* __restrict__ out, int n_elems) {
  int i = blockIdx.x * blockDim.x + threadIdx.x;
  int s = gridDim.x * blockDim.x;
  for (; i < n_elems; i += s) {
    int ch = i & (COUT - 1);
    float v = y[i] * ab[ch] + ab[COUT + ch];
    out[i] = xnet[i] + (v > 0.f ? v : 0.f);
  }
}

// ------------------------------------------------------------------- driver
static inline int cdiv(long a, long b) { return (int)((a + b - 1) / b); }

extern "C" void kernel_launch(void* const* d_in, const int* in_sizes, int n_in,
                              void* d_out, int out_size, void* d_ws, size_t ws_size,
                              hipStream_t stream) {
  const float* feats  = (const float*)d_in[0];
  const float* W_down = (const float*)d_in[1];
  const float* g0     = (const float*)d_in[2];
  const float* b0     = (const float*)d_in[3];
  const float* W1     = (const float*)d_in[4];
  const float* g1     = (const float*)d_in[5];
  const float* b1     = (const float*)d_in[6];
  const float* W2     = (const float*)d_in[7];
  const float* g2     = (const float*)d_in[8];
  const float* b2     = (const float*)d_in[9];
  const int* p_down_in  = (const int*)d_in[10];
  const int* p_down_out = (const int*)d_in[11];
  const int* p_subm_in  = (const int*)d_in[12];
  const int* p_subm_out = (const int*)d_in[13];

  const int Cin = 32;
  const int K   = in_sizes[1] / (Cin * COUT);   // 27
  const int P1  = in_sizes[10] / K;             // 100000
  const int P2  = in_sizes[12] / K;             // 30000
  const int n_out = out_size / COUT;            // 50000 (device scalar, derived)
  const int NE  = n_out * COUT;
  const float inv_n = 1.0f / (float)n_out;

  float* y     = (float*)d_ws;       // conv accumulator   [NE]
  float* x     = y + NE;             // bn_relu(conv1)     [NE]
  float* xnet  = x + NE;             // bn_relu(conv2)     [NE]
  float* stats = xnet + NE;          // [128]
  float* ab    = stats + 128;        // [128] scale/shift
  (void)n_in; (void)ws_size;

  const dim3 blk(256);
  const int zb  = cdiv(NE / 4, 256); // float4 zero-fill
  const int eb  = cdiv(NE, 256);
  const int sb  = 1024;              // grid-stride stats blocks

  // conv grids: (mtile-groups, K, ntiles)
  const int mt1 = cdiv(P1, 16);
  const int mt2 = cdiv(P2, 16);
  const dim3 cg1(cdiv(mt1, 8), K, 4);
  const dim3 cg2(cdiv(mt2, 8), K, 4);

  // ---- DownBlock: conv1 + BN + ReLU -> x
  zero_f32v4<<<zb, blk, 0, stream>>>((v4f*)y, NE / 4);
  zero_f32v4<<<1, 32, 0, stream>>>((v4f*)stats, 32);
  sparse_conv_wmma<32><<<cg1, blk, 0, stream>>>(feats, W_down, p_down_in, p_down_out,
                                                y, P1, mt1);
  col_stats<<<sb, blk, 0, stream>>>(y, NE, stats);
  bn_finalize<<<1, 64, 0, stream>>>(stats, g0, b0, inv_n, ab);
  bn_apply<<<eb, blk, 0, stream>>>(y, ab, x, NE);

  // ---- ResBlock conv2 + BN + ReLU -> xnet
  zero_f32v4<<<zb, blk, 0, stream>>>((v4f*)y, NE / 4);
  zero_f32v4<<<1, 32, 0, stream>>>((v4f*)stats, 32);
  sparse_conv_wmma<64><<<cg2, blk, 0, stream>>>(x, W1, p_subm_in, p_subm_out,
                                                y, P2, mt2);
  col_stats<<<sb, blk, 0, stream>>>(y, NE, stats);
  bn_finalize<<<1, 64, 0, stream>>>(stats, g1, b1, inv_n, ab);
  bn_apply<<<eb, blk, 0, stream>>>(y, ab, xnet, NE);

  // ---- ResBlock conv3 + BN + ReLU, fused residual add -> out
  zero_f32v4<<<zb, blk, 0, stream>>>((v4f*)y, NE / 4);
  zero_f32v4<<<1, 32, 0, stream>>>((v4f*)stats, 32);
  sparse_conv_wmma<64><<<cg2, blk, 0, stream>>>(xnet, W2, p_subm_in, p_subm_out,
                                                y, P2, mt2);
  col_stats<<<sb, blk, 0, stream>>>(y, NE, stats);
  bn_finalize<<<1, 64, 0, stream>>>(stats, g2, b2, inv_n, ab);
  bn_apply_add<<<eb, blk, 0, stream>>>(y, ab, xnet, (float*)d_out, NE);
}